// MuxGNN_24670292148300
// MI455X (gfx1250) — compile-verified
//
#include <hip/hip_runtime.h>

typedef float v2f __attribute__((ext_vector_type(2)));
typedef float v8f __attribute__((ext_vector_type(8)));

#define N_NODES 100000
#define N_EDGES 600000
#define IN_C    128
#define HID_C   128
#define OUT_C   64

// ---------------------------------------------------------------------------
// CSR build step 1: per-destination degree histogram (u32 atomics, cheap).
// ---------------------------------------------------------------------------
__global__ void __launch_bounds__(256)
count_kernel(const long long* __restrict__ ei, unsigned* __restrict__ deg, int n_edges) {
    int e = blockIdx.x * 256 + threadIdx.x;
    if (e >= n_edges) return;
    long long dst = ei[(long long)n_edges + e];
    atomicAdd(&deg[dst], 1u);
}

// ---------------------------------------------------------------------------
// CSR build step 2: exclusive prefix sum over N degrees -> row_ptr[N+1].
// Single 1024-thread block; wave-shuffle scans, 4 barriers per 1024 chunk.
// ---------------------------------------------------------------------------
__global__ void __launch_bounds__(1024)
scan_kernel(const unsigned* __restrict__ deg, unsigned* __restrict__ rp, int n) {
    __shared__ unsigned wsum[32];
    __shared__ unsigned carry, chunktot;
    const int t = threadIdx.x, lane = t & 31, wid = t >> 5;
    if (t == 0) carry = 0;
    __syncthreads();
    for (int base = 0; base < n; base += 1024) {
        int i = base + t;
        unsigned v = (i < n) ? deg[i] : 0u;
        unsigned s = v;
#pragma unroll
        for (int off = 1; off < 32; off <<= 1) {
            unsigned u = __shfl_up(s, off, 32);
            if (lane >= off) s += u;
        }
        if (lane == 31) wsum[wid] = s;
        __syncthreads();
        if (t < 32) {
            unsigned w = wsum[t];
            unsigned ss = w;
#pragma unroll
            for (int off = 1; off < 32; off <<= 1) {
                unsigned u = __shfl_up(ss, off, 32);
                if (t >= off) ss += u;
            }
            wsum[t] = ss - w;              // exclusive wave offset
            if (t == 31) chunktot = ss;    // chunk total
        }
        __syncthreads();
        if (i < n) rp[i] = carry + wsum[wid] + (s - v);
        __syncthreads();
        if (t == 0) carry += chunktot;
        __syncthreads();
    }
    if (t == 0) rp[n] = carry;
}

// CSR build step 3a: cursor[i] = rp[i]
__global__ void __launch_bounds__(256)
cursor_init_kernel(const unsigned* __restrict__ rp, unsigned* __restrict__ cur, int n) {
    int i = blockIdx.x * 256 + threadIdx.x;
    if (i < n) cur[i] = rp[i];
}

// CSR build step 3b: bucket-fill source indices.
__global__ void __launch_bounds__(256)
fill_kernel(const long long* __restrict__ ei, unsigned* __restrict__ cur,
            int* __restrict__ ci, int n_edges) {
    int e = blockIdx.x * 256 + threadIdx.x;
    if (e >= n_edges) return;
    long long src = ei[e];
    long long dst = ei[(long long)n_edges + e];
    unsigned pos = atomicAdd(&cur[dst], 1u);
    ci[pos] = (int)src;
}

// ---------------------------------------------------------------------------
// Weight repack into B-fragment order:
//   Wp[((c*32 + ks)*32 + lane)] = { W[4ks+kk][16c + (lane&15)],
//                                   W[4ks+kk+1][16c + (lane&15)] },  kk=(lane>>4)*2
// One coalesced global_load_b64 (256B/wave) per WMMA step instead of two
// scattered b32s.  nmat matrices of [128 x ncols], packed back-to-back.
// ---------------------------------------------------------------------------
__global__ void __launch_bounds__(256)
repack_kernel(const float* __restrict__ W, float* __restrict__ Wp,
              int ncols, int nmat) {
    const int per = (ncols / 16) * 32 * 32;      // float2 slots per matrix
    int tid = blockIdx.x * 256 + threadIdx.x;
    if (tid >= per * nmat) return;
    int m   = tid / per;
    int rix = tid - m * per;
    int lane = rix & 31;
    int ks   = (rix >> 5) & 31;
    int c    = rix >> 10;
    int kk   = (lane >> 4) << 1;
    int n    = c * 16 + (lane & 15);
    int k    = ks * 4 + kk;
    const float* w = W + (size_t)m * 128 * ncols;
    float2 v;
    v.x = w[(size_t)k * ncols + n];
    v.y = w[(size_t)(k + 1) * ncols + n];
    *(float2*)&Wp[(size_t)tid * 2] = v;
}

// ---------------------------------------------------------------------------
// Layer 1, fully fused over the 3 relations:
//   h[m,:] = (1/3) * sum_r relu( mean_agg_r(x)[m,:] @ Wl_r + bl_r + x[m,:] @ Wr_r )
// Block = 256 thr = 8 waves, one 16-row M-tile. Per relation: CSR gather of
// the agg tile into LDS (wave = 2 rows, one b128 load per edge), then K=128
// f32-WMMA GEMMs with fragment-packed weights; hsum in registers.
// ---------------------------------------------------------------------------
__global__ void __launch_bounds__(256)
sage_layer1_kernel(const float* __restrict__ x,
                   const unsigned* __restrict__ rp0, const int* __restrict__ ci0,
                   const unsigned* __restrict__ rp1, const int* __restrict__ ci1,
                   const unsigned* __restrict__ rp2, const int* __restrict__ ci2,
                   const float2* __restrict__ Wlp,  // packed [3][8][32][32] float2
                   const float* __restrict__ bl,    // [3][128]
                   const float2* __restrict__ Wrp,  // packed [3][8][32][32] float2
                   float* __restrict__ h) {
    __shared__ __align__(16) float xs[16 * 132];  // +4 pad rows
    __shared__ __align__(16) float as[16 * 132];

    const int m0   = blockIdx.x * 16;
    const int t    = threadIdx.x;
    const int wave = t >> 5;
    const int lane = t & 31;

    // stage x tile once (float4 / ds_store_b128)
#pragma unroll
    for (int i = 0; i < 2; ++i) {
        int slot = t + i * 256;                 // 512 float4 slots
        int row = slot >> 5, c4 = slot & 31;
        const float4 v = *(const float4*)(x + (size_t)(m0 + row) * IN_C + c4 * 4);
        *(float4*)&xs[row * 132 + c4 * 4] = v;
    }

    const int mr = lane & 15;
    const int nc = lane & 15;
    const int kk = (lane >> 4) << 1;
    const int n0 = wave * 16;
    const int rbase = ((lane >> 4) << 3);
    const int fragbase = wave * 32 * 32 + lane;   // float2 index of (c=wave, ks=0)

    v8f hsum = {};

#pragma unroll
    for (int r = 0; r < 3; ++r) {
        const unsigned* rp = (r == 0) ? rp0 : (r == 1) ? rp1 : rp2;
        const int*      ci = (r == 0) ? ci0 : (r == 1) ? ci1 : ci2;
        const float2*   wl = Wlp + (size_t)r * 8192 + fragbase;  // 8*32*32 f2/matrix
        const float2*   wr = Wrp + (size_t)r * 8192 + fragbase;

        __syncthreads();   // 'as' free (prev relation's GEMM done / first entry)

        // CSR gather: wave handles rows 2*wave, 2*wave+1
#pragma unroll
        for (int rr = 0; rr < 2; ++rr) {
            int row  = wave * 2 + rr;
            int node = m0 + row;
            unsigned s0 = rp[node], s1 = rp[node + 1];
            float a0 = 0.f, a1 = 0.f, a2 = 0.f, a3 = 0.f;
            for (unsigned e = s0; e < s1; ++e) {
                int src = ci[e];
                const float4 v = *(const float4*)(x + (size_t)src * IN_C + lane * 4);
                a0 += v.x; a1 += v.y; a2 += v.z; a3 += v.w;
            }
            float ri = 1.0f / fmaxf((float)(s1 - s0), 1.0f);
            float4 o; o.x = a0 * ri; o.y = a1 * ri; o.z = a2 * ri; o.w = a3 * ri;
            *(float4*)&as[row * 132 + lane * 4] = o;
        }
        __syncthreads();

        v8f acc = {};
#pragma unroll
        for (int ks = 0; ks < 32; ++ks) {       // agg_norm @ Wl_r
            int k0 = ks * 4;
            float2 bw = wl[ks * 32];
            v2f a, b;
            a.x = as[mr * 132 + k0 + kk];
            a.y = as[mr * 132 + k0 + kk + 1];
            b.x = bw.x; b.y = bw.y;
            acc = __builtin_amdgcn_wmma_f32_16x16x4_f32(false, a, false, b,
                                                        (short)0, acc, false, false);
        }
#pragma unroll
        for (int ks = 0; ks < 32; ++ks) {       // x @ Wr_r
            int k0 = ks * 4;
            float2 bw = wr[ks * 32];
            v2f a, b;
            a.x = xs[mr * 132 + k0 + kk];
            a.y = xs[mr * 132 + k0 + kk + 1];
            b.x = bw.x; b.y = bw.y;
            acc = __builtin_amdgcn_wmma_f32_16x16x4_f32(false, a, false, b,
                                                        (short)0, acc, false, false);
        }
        const float bias = bl[r * HID_C + n0 + nc];
#pragma unroll
        for (int j = 0; j < 8; ++j)
            hsum[j] += fmaxf(acc[j] + bias, 0.0f) * (1.0f / 3.0f);
    }

    // single write of h (D layout: VGPR j -> rows j / j+8)
#pragma unroll
    for (int j = 0; j < 8; ++j)
        h[(size_t)(m0 + rbase + j) * HID_C + n0 + nc] = hsum[j];
}

// ---------------------------------------------------------------------------
// Layer 2: out = mean_agg_0(h) @ Wl_f + bl_f + h @ Wr_f
// Block = 128 thr = 4 waves (64 cols), one 16-row M-tile; CSR-0 gather.
// ---------------------------------------------------------------------------
__global__ void __launch_bounds__(128)
sage_layer2_kernel(const float* __restrict__ h,
                   const unsigned* __restrict__ rp, const int* __restrict__ ci,
                   const float2* __restrict__ Wlfp,  // packed [4][32][32] float2
                   const float* __restrict__ bl_f,   // [64]
                   const float2* __restrict__ Wrfp,  // packed [4][32][32] float2
                   float* __restrict__ out) {
    __shared__ __align__(16) float hs[16 * 132];
    __shared__ __align__(16) float as[16 * 132];

    const int m0   = blockIdx.x * 16;
    const int t    = threadIdx.x;
    const int wave = t >> 5;
    const int lane = t & 31;

#pragma unroll
    for (int i = 0; i < 4; ++i) {
        int slot = t + i * 128;
        int row = slot >> 5, c4 = slot & 31;
        const float4 v = *(const float4*)(h + (size_t)(m0 + row) * HID_C + c4 * 4);
        *(float4*)&hs[row * 132 + c4 * 4] = v;
    }

    // CSR gather on h: wave handles rows 4*wave .. 4*wave+3
#pragma unroll
    for (int rr = 0; rr < 4; ++rr) {
        int row  = wave * 4 + rr;
        int node = m0 + row;
        unsigned s0 = rp[node], s1 = rp[node + 1];
        float a0 = 0.f, a1 = 0.f, a2 = 0.f, a3 = 0.f;
        for (unsigned e = s0; e < s1; ++e) {
            int src = ci[e];
            const float4 v = *(const float4*)(h + (size_t)src * HID_C + lane * 4);
            a0 += v.x; a1 += v.y; a2 += v.z; a3 += v.w;
        }
        float ri = 1.0f / fmaxf((float)(s1 - s0), 1.0f);
        float4 o; o.x = a0 * ri; o.y = a1 * ri; o.z = a2 * ri; o.w = a3 * ri;
        *(float4*)&as[row * 132 + lane * 4] = o;
    }
    __syncthreads();

    const int mr = lane & 15;
    const int nc = lane & 15;
    const int kk = (lane >> 4) << 1;
    const int n0 = wave * 16;
    const int fragbase = wave * 32 * 32 + lane;

    const float2* wl = Wlfp + fragbase;
    const float2* wr = Wrfp + fragbase;

    v8f acc = {};
#pragma unroll
    for (int ks = 0; ks < 32; ++ks) {
        int k0 = ks * 4;
        float2 bw = wl[ks * 32];
        v2f a, b;
        a.x = as[mr * 132 + k0 + kk];
        a.y = as[mr * 132 + k0 + kk + 1];
        b.x = bw.x; b.y = bw.y;
        acc = __builtin_amdgcn_wmma_f32_16x16x4_f32(false, a, false, b,
                                                    (short)0, acc, false, false);
    }
#pragma unroll
    for (int ks = 0; ks < 32; ++ks) {
        int k0 = ks * 4;
        float2 bw = wr[ks * 32];
        v2f a, b;
        a.x = hs[mr * 132 + k0 + kk];
        a.y = hs[mr * 132 + k0 + kk + 1];
        b.x = bw.x; b.y = bw.y;
        acc = __builtin_amdgcn_wmma_f32_16x16x4_f32(false, a, false, b,
                                                    (short)0, acc, false, false);
    }

    const float bias  = bl_f[n0 + nc];
    const int   rbase = m0 + ((lane >> 4) << 3);
#pragma unroll
    for (int j = 0; j < 8; ++j)
        out[(size_t)(rbase + j) * OUT_C + n0 + nc] = acc[j] + bias;
}

// ---------------------------------------------------------------------------
extern "C" void kernel_launch(void* const* d_in, const int* in_sizes, int n_in,
                              void* d_out, int out_size, void* d_ws, size_t ws_size,
                              hipStream_t stream) {
    const float*     x    = (const float*)d_in[0];
    const float*     Wl   = (const float*)d_in[1];
    const float*     bl   = (const float*)d_in[2];
    const float*     Wr   = (const float*)d_in[3];
    const float*     Wl_f = (const float*)d_in[4];
    const float*     bl_f = (const float*)d_in[5];
    const float*     Wr_f = (const float*)d_in[6];
    const long long* ei[3] = { (const long long*)d_in[7],
                               (const long long*)d_in[8],
                               (const long long*)d_in[9] };
    float* out = (float*)d_out;

    // workspace layout (~61.5 MB):
    //   h[N*128] | rp[3][N+1] | ci[3][E] | deg[N] | cur[N]
    //   | Wlp[3*16384] | Wrp[3*16384] | Wlfp[8192] | Wrfp[8192]
    float*    h   = (float*)d_ws;
    unsigned* rpb = (unsigned*)(h + (size_t)N_NODES * HID_C);
    unsigned* rp[3] = { rpb, rpb + (N_NODES + 1), rpb + 2 * (N_NODES + 1) };
    int*      cib = (int*)(rpb + 3 * (N_NODES + 1));
    int*      ci[3] = { cib, cib + N_EDGES, cib + 2 * (size_t)N_EDGES };
    unsigned* deg = (unsigned*)(cib + 3 * (size_t)N_EDGES);
    unsigned* cur = deg + N_NODES;
    float*    Wlp  = (float*)(cur + N_NODES);
    float*    Wrp  = Wlp + 3 * (size_t)IN_C * HID_C;
    float*    Wlfp = Wrp + 3 * (size_t)IN_C * HID_C;
    float*    Wrfp = Wlfp + (size_t)HID_C * OUT_C;

    const dim3 egrid((N_EDGES + 255) / 256);
    const dim3 ngrid((N_NODES + 255) / 256);
    const dim3 mgrid(N_NODES / 16);          // 100000 = 6250 * 16 exactly

    // ---- repack weights into B-fragment order (tiny) ----
    repack_kernel<<<(3 * 8192 + 255) / 256, 256, 0, stream>>>(Wl, Wlp, HID_C, 3);
    repack_kernel<<<(3 * 8192 + 255) / 256, 256, 0, stream>>>(Wr, Wrp, HID_C, 3);
    repack_kernel<<<(4096 + 255) / 256, 256, 0, stream>>>(Wl_f, Wlfp, OUT_C, 1);
    repack_kernel<<<(4096 + 255) / 256, 256, 0, stream>>>(Wr_f, Wrfp, OUT_C, 1);

    // ---- build CSR per relation (dst-bucketed) ----
    for (int r = 0; r < 3; ++r) {
        hipMemsetAsync(deg, 0, (size_t)N_NODES * sizeof(unsigned), stream);
        count_kernel<<<egrid, 256, 0, stream>>>(ei[r], deg, N_EDGES);
        scan_kernel<<<1, 1024, 0, stream>>>(deg, rp[r], N_NODES);
        cursor_init_kernel<<<ngrid, 256, 0, stream>>>(rp[r], cur, N_NODES);
        fill_kernel<<<egrid, 256, 0, stream>>>(ei[r], cur, ci[r], N_EDGES);
    }

    // ---- fused layer 1 (all 3 relations, gather + WMMA, single h write) ----
    sage_layer1_kernel<<<mgrid, 256, 0, stream>>>(
        x, rp[0], ci[0], rp[1], ci[1], rp[2], ci[2],
        (const float2*)Wlp, bl, (const float2*)Wrp, h);

    // ---- layer 2 (reuses relation-0 CSR) ----
    sage_layer2_kernel<<<mgrid, 128, 0, stream>>>(
        h, rp[0], ci[0], (const float2*)Wlfp, bl_f, (const float2*)Wrfp, out);
}